// PrototypeClassifier_25237227831300
// MI455X (gfx1250) — compile-verified
//
#include <hip/hip_runtime.h>
#include <math.h>

#define NUM_CLASSES 1000
#define NPC 8
#define HIDDEN 1024
#define BATCH 512
#define NPROTO (NUM_CLASSES * NPC)
#define MARGIN_F (-0.1f)
#define LOSS_WEIGHT_F 0.1f

typedef __attribute__((ext_vector_type(2))) float v2f;
typedef __attribute__((ext_vector_type(8))) float v8f;

// ---------------- zero fill ----------------
__global__ void k_zero(float* __restrict__ a, int n) {
    int i = blockIdx.x * 256 + threadIdx.x;
    if (i < n) a[i] = 0.0f;
}

// ---------------- x2[b] = sum_h hidden[b,h]^2 ----------------
__global__ void __launch_bounds__(256) k_rowsq(const float* __restrict__ x,
                                               float* __restrict__ out) {
    __shared__ float red[256];
    const int row = blockIdx.x;
    const float* r = x + (size_t)row * HIDDEN;
    float s = 0.0f;
    for (int c = threadIdx.x; c < HIDDEN; c += 256) { float v = r[c]; s += v * v; }
    red[threadIdx.x] = s;
    __syncthreads();
    for (int o = 128; o > 0; o >>= 1) {
        if (threadIdx.x < o) red[threadIdx.x] += red[threadIdx.x + o];
        __syncthreads();
    }
    if (threadIdx.x == 0) out[row] = red[0];
}

// ------- p2[p] = ||proto_p||^2 ; svec += proto_p / ||proto_p||  (64 protos/block) -------
__global__ void __launch_bounds__(256) k_proto(const float* __restrict__ proto,
                                               float* __restrict__ p2,
                                               float* __restrict__ svec) {
    __shared__ float red[256];
    __shared__ float rn_sh;
    const int p0 = blockIdx.x * 64;
    float acc0 = 0.f, acc1 = 0.f, acc2 = 0.f, acc3 = 0.f;
    for (int i = 0; i < 64; ++i) {
        const int p = p0 + i;
        const float* r = proto + (size_t)p * HIDDEN;
        float v0 = r[threadIdx.x +   0];
        float v1 = r[threadIdx.x + 256];
        float v2 = r[threadIdx.x + 512];
        float v3 = r[threadIdx.x + 768];
        red[threadIdx.x] = v0 * v0 + v1 * v1 + v2 * v2 + v3 * v3;
        __syncthreads();
        for (int o = 128; o > 0; o >>= 1) {
            if (threadIdx.x < o) red[threadIdx.x] += red[threadIdx.x + o];
            __syncthreads();
        }
        if (threadIdx.x == 0) {
            p2[p] = red[0];
            rn_sh = 1.0f / sqrtf(red[0]);
        }
        __syncthreads();
        const float rn = rn_sh;
        acc0 += v0 * rn; acc1 += v1 * rn; acc2 += v2 * rn; acc3 += v3 * rn;
        __syncthreads();   // protect red[0]/rn_sh before next iteration rewrites them
    }
    atomicAdd(&svec[threadIdx.x +   0], acc0);
    atomicAdd(&svec[threadIdx.x + 256], acc1);
    atomicAdd(&svec[threadIdx.x + 512], acc2);
    atomicAdd(&svec[threadIdx.x + 768], acc3);
}

// distance + per-class (NPC=8) max/argmax epilogue for one 16x16 C tile
__device__ __forceinline__ void tile_epilogue(const v8f& acc, int r0, int pt,
                                              int hi, int ln,
                                              const float* __restrict__ x2,
                                              const float* __restrict__ p2,
                                              float* __restrict__ out,
                                              int* __restrict__ topk) {
    const float pp = p2[pt + ln];
#pragma unroll
    for (int v = 0; v < 8; ++v) {
        const int b = r0 + v + 8 * hi;                   // batch row for this C slot
        const float d2 = x2[b] + pp - 2.0f * acc[v];
        float val = -sqrtf(fmaxf(d2, 0.0f));             // sim
        int   idx = ln & 7;                              // prototype index within class
        // max+argmax over the 8 lanes of one class (first-index tie-break, as jnp.argmax)
#pragma unroll
        for (int off = 1; off < 8; off <<= 1) {
            float oval = __shfl_xor(val, off, 32);
            int   oidx = __shfl_xor(idx, off, 32);
            if (oval > val || (oval == val && oidx < idx)) { val = oval; idx = oidx; }
        }
        if ((ln & 7) == 0) {
            const int cls = (pt >> 3) + (ln >> 3);
            out [(size_t)b * NUM_CLASSES + cls] = val;
            topk[(size_t)b * NUM_CLASSES + cls] = idx;
        }
    }
}

// ---------------- fp32 WMMA GEMM + distance epilogue ----------------
// grid = (8000/64, 512/64); block = 128 (4 waves, 2x2 over a 64x64 block tile).
// Each wave: 32 batch rows x 32 prototypes = four 16x16 accumulators (independent
// WMMA dependency chains; 4 loads feed 4 wmma per k-step -> 8 FLOP/byte from L1).
__global__ void __launch_bounds__(128) k_gemm(const float* __restrict__ hidden,
                                              const float* __restrict__ proto,
                                              const float* __restrict__ x2,
                                              const float* __restrict__ p2,
                                              float* __restrict__ out,
                                              int* __restrict__ topk) {
    const int lane = threadIdx.x & 31;
    const int wave = threadIdx.x >> 5;
    const int hi   = lane >> 4;       // 0: K-pair {0,1}, 1: K-pair {2,3}
    const int ln   = lane & 15;       // A: row M ; B: col N (prototype)
    const int wrow = wave >> 1;       // 0..1
    const int wcol = wave & 1;        // 0..1
    const int row0  = blockIdx.y * 64 + wrow * 32;
    const int ptile = blockIdx.x * 64 + wcol * 32;

    const float* a0 = hidden + (size_t)(row0  + ln) * HIDDEN + 2 * hi;
    const float* a1 = a0 + (size_t)16 * HIDDEN;
    const float* b0 = proto  + (size_t)(ptile + ln) * HIDDEN + 2 * hi;
    const float* b1 = b0 + (size_t)16 * HIDDEN;

    v8f acc00 = {}, acc01 = {}, acc10 = {}, acc11 = {};
#pragma unroll 4
    for (int k = 0; k < HIDDEN; k += 4) {
        v2f A0 = *reinterpret_cast<const v2f*>(a0 + k);
        v2f A1 = *reinterpret_cast<const v2f*>(a1 + k);
        v2f B0 = *reinterpret_cast<const v2f*>(b0 + k);
        v2f B1 = *reinterpret_cast<const v2f*>(b1 + k);
        acc00 = __builtin_amdgcn_wmma_f32_16x16x4_f32(false, A0, false, B0,
                                                      (short)0, acc00, false, false);
        acc01 = __builtin_amdgcn_wmma_f32_16x16x4_f32(false, A0, false, B1,
                                                      (short)0, acc01, false, false);
        acc10 = __builtin_amdgcn_wmma_f32_16x16x4_f32(false, A1, false, B0,
                                                      (short)0, acc10, false, false);
        acc11 = __builtin_amdgcn_wmma_f32_16x16x4_f32(false, A1, false, B1,
                                                      (short)0, acc11, false, false);
    }

    tile_epilogue(acc00, row0,      ptile,      hi, ln, x2, p2, out, topk);
    tile_epilogue(acc01, row0,      ptile + 16, hi, ln, x2, p2, out, topk);
    tile_epilogue(acc10, row0 + 16, ptile,      hi, ln, x2, p2, out, topk);
    tile_epilogue(acc11, row0 + 16, ptile + 16, hi, ln, x2, p2, out, topk);
}

// ---------------- per-row: argmax, softmax CE, margin mask, usage counters ----------------
__global__ void __launch_bounds__(256) k_rowloss(const float* __restrict__ out,
                                                 const int* __restrict__ topk,
                                                 const int* __restrict__ labels,
                                                 float* __restrict__ scal,       // [0]=n_kept [1]=sum_ce
                                                 float* __restrict__ pos_usage,
                                                 float* __restrict__ neg_usage) {
    __shared__ float rv[256];
    __shared__ int   ri[256];
    const int b = blockIdx.x;
    const float* row = out + (size_t)b * NUM_CLASSES;

    float best = -__builtin_inff();
    int   bidx = 0x7fffffff;
    for (int c = threadIdx.x; c < NUM_CLASSES; c += 256) {
        float v = row[c];
        if (v > best) { best = v; bidx = c; }            // strict > keeps first index
    }
    rv[threadIdx.x] = best; ri[threadIdx.x] = bidx;
    __syncthreads();
    for (int o = 128; o > 0; o >>= 1) {
        if (threadIdx.x < o) {
            float ov = rv[threadIdx.x + o]; int oi = ri[threadIdx.x + o];
            if (ov > rv[threadIdx.x] || (ov == rv[threadIdx.x] && oi < ri[threadIdx.x])) {
                rv[threadIdx.x] = ov; ri[threadIdx.x] = oi;
            }
        }
        __syncthreads();
    }
    const float rowmax = rv[0];
    const int   pred   = ri[0];
    __syncthreads();

    float se = 0.0f;
    for (int c = threadIdx.x; c < NUM_CLASSES; c += 256) se += expf(row[c] - rowmax);
    rv[threadIdx.x] = se;
    __syncthreads();
    for (int o = 128; o > 0; o >>= 1) {
        if (threadIdx.x < o) rv[threadIdx.x] += rv[threadIdx.x + o];
        __syncthreads();
    }

    if (threadIdx.x == 0) {
        const float sumexp  = rv[0];
        const int   label   = labels[b];
        const float correct = row[label];
        const float margin  = correct - rowmax;          // predicted_score == rowmax
        if (margin > MARGIN_F) {
            const float per_ce = logf(sumexp) + rowmax - correct;  // -log_softmax[label]
            atomicAdd(&scal[0], 1.0f);
            atomicAdd(&scal[1], per_ce);
        }
        atomicAdd(&pos_usage[label * NPC + topk[(size_t)b * NUM_CLASSES + label]], 1.0f);
        if (pred != label)
            atomicAdd(&neg_usage[pred * NPC + topk[(size_t)b * NUM_CLASSES + pred]], 1.0f);
    }
}

// ---------------- finalize total loss ----------------
// dispersion_loss = -mean(1 - N N^T) = ||sum_i n_i||^2 / P^2 - 1
__global__ void __launch_bounds__(256) k_final(const float* __restrict__ svec,
                                               const float* __restrict__ scal,
                                               float* __restrict__ loss_out) {
    __shared__ float red[256];
    float s = 0.0f;
    for (int i = threadIdx.x; i < HIDDEN; i += 256) { float v = svec[i]; s += v * v; }
    red[threadIdx.x] = s;
    __syncthreads();
    for (int o = 128; o > 0; o >>= 1) {
        if (threadIdx.x < o) red[threadIdx.x] += red[threadIdx.x + o];
        __syncthreads();
    }
    if (threadIdx.x == 0) {
        const float ss   = red[0];
        const float disp = ss / ((float)NPROTO * (float)NPROTO) - 1.0f;
        const float nk   = scal[0];
        const float total = (nk > 0.0f)
            ? (scal[1] / fmaxf(nk, 1.0f) + LOSS_WEIGHT_F * disp)
            : 0.0f;
        *loss_out = total;
    }
}

extern "C" void kernel_launch(void* const* d_in, const int* in_sizes, int n_in,
                              void* d_out, int out_size, void* d_ws, size_t ws_size,
                              hipStream_t stream) {
    const float* hidden = (const float*)d_in[0];   // [512,1024]
    const float* proto  = (const float*)d_in[1];   // [8000,1024]
    const int*   labels = (const int*)d_in[2];     // [512]

    float* out  = (float*)d_out;                   // [512,1000]
    float* loss = out + (size_t)BATCH * NUM_CLASSES;        // 1
    float* posu = loss + 1;                                 // [8000]
    float* negu = posu + NPROTO;                            // [8000]

    float* ws   = (float*)d_ws;
    int*   topk = (int*)ws;                                 // [512,1000] int32
    float* x2   = ws + (size_t)BATCH * NUM_CLASSES;         // [512]
    float* p2   = x2 + BATCH;                               // [8000]
    float* svec = p2 + NPROTO;                              // [1024]
    float* scal = svec + HIDDEN;                            // [2]: n_kept, sum_ce

    // zero accumulators (svec+scal contiguous) and usage counters (posu+negu contiguous)
    k_zero<<<(HIDDEN + 2 + 255) / 256, 256, 0, stream>>>(svec, HIDDEN + 2);
    k_zero<<<(2 * NPROTO + 255) / 256, 256, 0, stream>>>(posu, 2 * NPROTO);

    k_rowsq<<<BATCH, 256, 0, stream>>>(hidden, x2);
    k_proto<<<NPROTO / 64, 256, 0, stream>>>(proto, p2, svec);

    k_gemm<<<dim3(NPROTO / 64, BATCH / 64), 128, 0, stream>>>(hidden, proto, x2, p2, out, topk);

    k_rowloss<<<BATCH, 256, 0, stream>>>(out, topk, labels, scal, posu, negu);
    k_final<<<1, 256, 0, stream>>>(svec, scal, loss);
}